// GCN_24893630448162
// MI455X (gfx1250) — compile-verified
//
#include <hip/hip_runtime.h>
#include <math.h>

// Problem constants (match reference)
#define NN 131072      // nodes
#define EE 1048576     // edges
#define DD 128         // emb size
#define HH 128         // hidden
#define TT 4           // edge types
#define LL 2           // layers
#define GB 512         // graphs
#define PPG 32         // predicate nodes per graph
#define KP_PROJ 136    // 133 padded up to multiple of 8
#define KROWS_PROJ 133
#define WSTRIDE 288    // floats per k-pair row in LDS (288 mod 64 == 32 -> no bank conflicts)

typedef __attribute__((ext_vector_type(2))) float v2f;
typedef __attribute__((ext_vector_type(8))) float v8f;

#define AS1 __attribute__((address_space(1)))
#define AS3 __attribute__((address_space(3)))

#if __has_builtin(__builtin_amdgcn_global_load_async_to_lds_b32)
#define HAVE_ASYNC_LDS 1
#else
#define HAVE_ASYNC_LDS 0
#endif

__device__ __forceinline__ void wait_asynccnt0() {
#if HAVE_ASYNC_LDS
#if __has_builtin(__builtin_amdgcn_s_wait_asynccnt)
  __builtin_amdgcn_s_wait_asynccnt(0);
#else
  asm volatile("s_wait_asynccnt 0" ::: "memory");
#endif
#endif
}

// ---------------------------------------------------------------------------
// init_forward: ne-edge aggregation  agg0[dst] += emb_table[gid[src]] * norm
// ---------------------------------------------------------------------------
__global__ void k_ne_edge(const float* __restrict__ emb_table,
                          const int* __restrict__ global_id,
                          const int* __restrict__ src,
                          const int* __restrict__ dst,
                          const int* __restrict__ etype,
                          const float* __restrict__ norm,
                          float* __restrict__ agg,
                          float* __restrict__ deg) {
  int e = blockIdx.x * 8 + (threadIdx.x >> 5);
  int c = (threadIdx.x & 31) * 4;
  if (etype[e] != 0) return;  // 'ne' == 0
  int s = src[e], d = dst[e];
  float nv = norm[e];
  const float* row = emb_table + (size_t)global_id[s] * DD + c;
  float* o = agg + (size_t)d * DD + c;
  atomicAdd(o + 0, row[0] * nv);
  atomicAdd(o + 1, row[1] * nv);
  atomicAdd(o + 2, row[2] * nv);
  atomicAdd(o + 3, row[3] * nv);
  if (c == 0) atomicAdd(&deg[d], 1.0f);
}

// ---------------------------------------------------------------------------
// Build padded feature matrix Xp[N][136]
// ---------------------------------------------------------------------------
__global__ void k_build_x(const float* __restrict__ emb_table,
                          const int* __restrict__ global_id,
                          const int* __restrict__ spo,
                          const int* __restrict__ access,
                          const int* __restrict__ pre_access,
                          const float* __restrict__ agg,
                          const float* __restrict__ deg,
                          float* __restrict__ Xp) {
  size_t idx = (size_t)blockIdx.x * blockDim.x + threadIdx.x;
  if (idx >= (size_t)NN * KP_PROJ) return;
  int n = (int)(idx / KP_PROJ);
  int c = (int)(idx % KP_PROJ);
  float val;
  if (c < DD) {
    bool is_ne = (spo[n * 3 + 0] + spo[n * 3 + 2]) > 0;
    bool has   = deg[n] > 0.5f;
    val = (is_ne && has) ? agg[(size_t)n * DD + c]
                         : emb_table[(size_t)global_id[n] * DD + c];
  } else if (c < 131) {
    val = (float)spo[n * 3 + (c - 128)];
  } else if (c == 131) {
    val = (float)access[n];
  } else if (c == 132) {
    val = (float)pre_access[n];
  } else {
    val = 0.0f;
  }
  Xp[idx] = val;
}

// ---------------------------------------------------------------------------
// Dense GEMM  C[N][128] = A[N][Kpad(lda)] @ W[Krows][128] (+ bias)
// V_WMMA_F32_16X16X4_F32; 256 threads = 8 waves; wave owns a 16x128 strip.
// W staged in LDS via CDNA5 async global->LDS engine; k-pair-interleaved
// layout with WSTRIDE padding (bank-conflict free).  Inner loop is manually
// software-pipelined: B fragments double-buffered (bcur/bnext) and A global
// loads prefetched one iteration ahead, so each 8-WMMA burst hides the
// LDS/global latency of the next burst's operands.
//
// Fragment layout (f32, 16x16x4):
//   A: lane L<16 holds A[m0+L][k0+0..1], lane L>=16 holds A[m0+L-16][k0+2..3]
//   B: lane L<16 holds W[k0+0..1][n0+L], lane L>=16 holds W[k0+2..3][n0+L-16]
//   C/D: vgpr r, lane L: row m0 + r + 8*(L>>4), col n0 + (L&15)
// ---------------------------------------------------------------------------
__global__ void __launch_bounds__(256)
k_gemm_wmma(const float* __restrict__ A, int lda, int Kpad,
            const float* __restrict__ W, int Krows,
            const float* __restrict__ bias,
            float* __restrict__ C) {
  extern __shared__ float lds_w[];  // (Kpad/2 + 1)*WSTRIDE floats

  // ---- stage W into LDS (interleaved pairs, padded stride) ----
#if HAVE_ASYNC_LDS
  for (int i = threadIdx.x; i < Krows * HH; i += 256) {
    int k = i >> 7;
    int n = i & 127;
    int off = (k >> 1) * WSTRIDE + n * 2 + (k & 1);
    __builtin_amdgcn_global_load_async_to_lds_b32(
        (AS1 int*)(W + i), (AS3 int*)(&lds_w[off]), 0, 0);
  }
  for (int i = Krows * HH + threadIdx.x; i < Kpad * HH; i += 256) {
    int k = i >> 7;
    int n = i & 127;
    lds_w[(k >> 1) * WSTRIDE + n * 2 + (k & 1)] = 0.0f;
  }
  wait_asynccnt0();
#else
  for (int i = threadIdx.x; i < Kpad * HH; i += 256) {
    int k = i >> 7;
    int n = i & 127;
    float w = (k < Krows) ? W[(size_t)k * HH + n] : 0.0f;
    lds_w[(k >> 1) * WSTRIDE + n * 2 + (k & 1)] = w;
  }
#endif
  __syncthreads();

  int lane  = threadIdx.x & 31;
  int wave  = threadIdx.x >> 5;
  int m0    = blockIdx.x * 128 + wave * 16;
  int mrow  = m0 + (lane & 15);
  int khalf = (lane >> 4) * 2;   // 0 or 2
  int col16 = lane & 15;

  v8f acc[8] = {};

  const float* arow = A + (size_t)mrow * lda + khalf;
  const float* bcol = lds_w + (khalf >> 1) * WSTRIDE + col16 * 2;

  // ---- software-pipelined main loop (2 k-steps per iteration) ----
  v2f bcur[8], bnext[8];
#pragma unroll
  for (int j = 0; j < 8; ++j) bcur[j] = *(const v2f*)(bcol + j * 32);
  v2f a0p = *(const v2f*)(arow + 0);
  v2f a1p = *(const v2f*)(arow + 4);

  for (int k0 = 0; k0 < Kpad; k0 += 8) {
    v2f a0 = a0p;
    v2f a1 = a1p;
    // prefetch next iteration's A (overreads stay inside the workspace)
    a0p = *(const v2f*)(arow + k0 + 8);
    a1p = *(const v2f*)(arow + k0 + 12);

    // load B fragments for k0+4 while WMMAs consume bcur (k0)
    const float* bb1 = bcol + ((k0 + 4) >> 1) * WSTRIDE;
#pragma unroll
    for (int j = 0; j < 8; ++j) bnext[j] = *(const v2f*)(bb1 + j * 32);
#pragma unroll
    for (int j = 0; j < 8; ++j) {
      acc[j] = __builtin_amdgcn_wmma_f32_16x16x4_f32(
          false, a0, false, bcur[j], (short)0, acc[j], false, false);
    }

    // load B fragments for k0+8 (extra LDS pad row covers final overread)
    const float* bb2 = bcol + ((k0 + 8) >> 1) * WSTRIDE;
#pragma unroll
    for (int j = 0; j < 8; ++j) bcur[j] = *(const v2f*)(bb2 + j * 32);
#pragma unroll
    for (int j = 0; j < 8; ++j) {
      acc[j] = __builtin_amdgcn_wmma_f32_16x16x4_f32(
          false, a1, false, bnext[j], (short)0, acc[j], false, false);
    }
  }

  // ---- store (optionally fused bias add) ----
  int rbase = (lane >> 4) * 8;
#pragma unroll
  for (int j = 0; j < 8; ++j) {
    int coln = (j << 4) + col16;
    float badd = bias ? bias[coln] : 0.0f;
#pragma unroll
    for (int r = 0; r < 8; ++r) {
      int row = m0 + rbase + r;
      C[(size_t)row * HH + coln] = acc[j][r] + badd;
    }
  }
}

// ---------------------------------------------------------------------------
// RGCN message scatter for one edge type: agg[dst] += mt[src] * norm
// ---------------------------------------------------------------------------
__global__ void k_edge_scatter(const float* __restrict__ mt,
                               const int* __restrict__ src,
                               const int* __restrict__ dst,
                               const int* __restrict__ etype,
                               const float* __restrict__ norm,
                               float* __restrict__ agg,
                               int t) {
  int e = blockIdx.x * 8 + (threadIdx.x >> 5);
  int c = (threadIdx.x & 31) * 4;
  if (etype[e] != t) return;
  int s = src[e], d = dst[e];
  float nv = norm[e];
  const float* row = mt + (size_t)s * HH + c;
  float* o = agg + (size_t)d * HH + c;
  atomicAdd(o + 0, row[0] * nv);
  atomicAdd(o + 1, row[1] * nv);
  atomicAdd(o + 2, row[2] * nv);
  atomicAdd(o + 3, row[3] * nv);
}

// ---------------------------------------------------------------------------
// h = relu(agg + b_r[l]) (+ h residual when residual != 0; in-place safe)
// ---------------------------------------------------------------------------
__global__ void k_combine(float* __restrict__ h,
                          const float* __restrict__ agg,
                          const float* __restrict__ brow,
                          int residual) {
  size_t idx = (size_t)blockIdx.x * blockDim.x + threadIdx.x;
  if (idx >= (size_t)NN * HH) return;
  int c = (int)(idx & 127);
  float v = agg[idx] + brow[c];
  v = v > 0.0f ? v : 0.0f;
  h[idx] = residual ? (v + h[idx]) : v;
}

// ---------------------------------------------------------------------------
// Predict: per-graph pooling + bilinear + log_softmax.  One block per graph.
// ---------------------------------------------------------------------------
__global__ void __launch_bounds__(128)
k_predict(const float* __restrict__ h,
          const int* __restrict__ pred_idx,
          const float* __restrict__ Wb,
          const float* __restrict__ bbp,
          float* __restrict__ out) {
  __shared__ float hp[PPG * HH];
  __shared__ float poolS[HH];
  __shared__ float vS[HH];
  __shared__ float outS[PPG];
  __shared__ float lseS;

  int b = blockIdx.x;
  int tid = threadIdx.x;

  for (int i = tid; i < PPG * HH; i += 128) {
    int p = i >> 7;
    int c = i & 127;
    int node = pred_idx[b * PPG + p];
    hp[i] = h[(size_t)node * HH + c];
  }
  __syncthreads();

  {
    float s = 0.0f;
    for (int p = 0; p < PPG; ++p) s += hp[p * HH + tid];
    poolS[tid] = s * (1.0f / PPG);
  }
  __syncthreads();

  {
    float s = 0.0f;
    const float* wr = Wb + (size_t)tid * HH;
    for (int k = 0; k < HH; ++k) s += wr[k] * poolS[k];
    vS[tid] = s;
  }
  __syncthreads();

  if (tid < PPG) {
    float s = bbp[0];
    const float* hr = &hp[tid * HH];
    for (int c = 0; c < HH; ++c) s += hr[c] * vS[c];
    outS[tid] = s;
  }
  __syncthreads();

  if (tid == 0) {
    float m = -3.4e38f;
    for (int p = 0; p < PPG; ++p) m = fmaxf(m, outS[p]);
    float se = 0.0f;
    for (int p = 0; p < PPG; ++p) se += expf(outS[p] - m);
    lseS = m + logf(se);
  }
  __syncthreads();

  if (tid < PPG) out[b * PPG + tid] = outS[tid] - lseS;
}

// ---------------------------------------------------------------------------
extern "C" void kernel_launch(void* const* d_in, const int* in_sizes, int n_in,
                              void* d_out, int out_size, void* d_ws, size_t ws_size,
                              hipStream_t stream) {
  const float* emb_table  = (const float*)d_in[0];   // V*D
  const float* W_t        = (const float*)d_in[1];   // 133*128
  const float* b_t        = (const float*)d_in[2];   // 128
  const float* W_r        = (const float*)d_in[3];   // L*T*H*H
  const float* b_r        = (const float*)d_in[4];   // L*H
  const float* Wb         = (const float*)d_in[5];   // H*H
  const float* bb         = (const float*)d_in[6];   // 1
  const float* norm       = (const float*)d_in[7];   // E
  const int*   global_id  = (const int*)d_in[8];     // N
  const int*   spo        = (const int*)d_in[9];     // N*3
  const int*   access     = (const int*)d_in[10];    // N
  const int*   pre_access = (const int*)d_in[11];    // N
  const int*   srcI       = (const int*)d_in[12];    // E
  const int*   dstI       = (const int*)d_in[13];    // E
  const int*   etype      = (const int*)d_in[14];    // E
  const int*   pred_idx   = (const int*)d_in[15];    // B*PPG

  float* ws  = (float*)d_ws;
  float* Xp  = ws;                                   // N*136 (aliased as mt later)
  float* mt  = ws;                                   // alias: Xp dead after projection
  float* h   = ws + (size_t)NN * KP_PROJ;            // N*128
  float* agg = h + (size_t)NN * HH;                  // N*128
  float* deg = agg + (size_t)NN * HH;                // N

  // ---- init_forward ----
  (void)hipMemsetAsync(agg, 0, (size_t)NN * HH * sizeof(float), stream);
  (void)hipMemsetAsync(deg, 0, (size_t)NN * sizeof(float), stream);
  k_ne_edge<<<EE / 8, 256, 0, stream>>>(emb_table, global_id, srcI, dstI,
                                        etype, norm, agg, deg);
  k_build_x<<<((size_t)NN * KP_PROJ + 255) / 256, 256, 0, stream>>>(
      emb_table, global_id, spo, access, pre_access, agg, deg, Xp);

  // ---- input projection: h = Xp @ W_t + b_t (WMMA) ----
  k_gemm_wmma<<<NN / 128, 256,
                (KP_PROJ / 2 + 1) * WSTRIDE * sizeof(float), stream>>>(
      Xp, KP_PROJ, KP_PROJ, W_t, KROWS_PROJ, b_t, h);

  // ---- RGCN layers ----
  for (int l = 0; l < LL; ++l) {
    (void)hipMemsetAsync(agg, 0, (size_t)NN * HH * sizeof(float), stream);
    for (int t = 0; t < TT; ++t) {
      const float* Wlt = W_r + ((size_t)(l * TT + t)) * HH * HH;
      k_gemm_wmma<<<NN / 128, 256,
                    (HH / 2 + 1) * WSTRIDE * sizeof(float), stream>>>(
          h, HH, HH, Wlt, HH, nullptr, mt);
      k_edge_scatter<<<EE / 8, 256, 0, stream>>>(mt, srcI, dstI, etype, norm,
                                                 agg, t);
    }
    k_combine<<<((size_t)NN * HH + 255) / 256, 256, 0, stream>>>(
        h, agg, b_r + l * HH, (l == 0) ? 1 : 0);
  }

  // ---- predict ----
  k_predict<<<GB, 128, 0, stream>>>(h, pred_idx, Wb, bb, (float*)d_out);
}